// AI4DEM_12592844112255
// MI455X (gfx1250) — compile-verified
//
#include <hip/hip_runtime.h>
#include <math.h>
#include <stdint.h>

#define NN 192
#define TDX 16
#define TDY 4
#define TDZ 4
#define HXD (TDX + 4)              // 20
#define HYD (TDY + 4)              // 8
#define HZD (TDZ + 4)              // 8
#define HVOL (HXD * HYD * HZD)     // 1280
#define NTHREADS (TDX * TDY * TDZ) // 256

#define KN_C 500000.0f
#define MU_C 0.5f
#define EPS_C 1e-4f

__device__ __forceinline__ int wrapN(int v) {
  // v is in [-2, NN+1]
  if (v < 0) v += NN;
  else if (v >= NN) v -= NN;
  return v;
}

__global__ __launch_bounds__(NTHREADS)
void dem_forces_kernel(const float* __restrict__ xg, const float* __restrict__ yg,
                       const float* __restrict__ zg, const float* __restrict__ vxg,
                       const float* __restrict__ vyg, const float* __restrict__ vzg,
                       const float* __restrict__ dptr, float* __restrict__ out,
                       float eta) {
  __shared__ float sh[6][HZD][HYD][HXD]; // 30720 bytes

  const int tx = threadIdx.x, ty = threadIdx.y, tz = threadIdx.z;
  const int tid = tx + TDX * (ty + TDY * tz);
  const int bx = blockIdx.x * TDX, by = blockIdx.y * TDY, bz = blockIdx.z * TDZ;

  const float* srcs[6] = {xg, yg, zg, vxg, vyg, vzg};

  // ---- async halo fill: global -> LDS, tracked by ASYNCcnt (CDNA5 path) ----
  #pragma unroll
  for (int g = 0; g < 6; ++g) {
    const float* src = srcs[g];
    #pragma unroll
    for (int r = 0; r < HVOL / NTHREADS; ++r) { // exactly 5 per thread
      const int h  = tid + r * NTHREADS;
      const int hx = h % HXD;
      const int t1 = h / HXD;
      const int hy = t1 % HYD;
      const int hz = t1 / HYD;
      const int gx = wrapN(bx + hx - 2);
      const int gy = wrapN(by + hy - 2);
      const int gz = wrapN(bz + hz - 2);
      // low 32 bits of a generic pointer to LDS == LDS byte offset
      const unsigned lds_off = (unsigned)(uintptr_t)&sh[g][hz][hy][hx];
      const unsigned long long gaddr =
          (unsigned long long)(uintptr_t)(src + ((size_t)gz * NN + gy) * NN + gx);
      asm volatile("global_load_async_to_lds_b32 %0, %1, off"
                   :: "v"(lds_off), "v"(gaddr)
                   : "memory");
    }
  }
  asm volatile("s_wait_asynccnt 0x0" ::: "memory");
  __syncthreads();

  // ---- per-cell accumulation over 125 shifts (reference order: i,j,k) ----
  const int cx = tx + 2, cy = ty + 2, cz = tz + 2;
  const float d     = dptr[0];
  const float two_d = 2.0f * d;

  const float x0  = sh[0][cz][cy][cx];
  const float y0  = sh[1][cz][cy][cx];
  const float z0  = sh[2][cz][cy][cx];
  const float vx0 = sh[3][cz][cy][cx];
  const float vy0 = sh[4][cz][cy][cx];
  const float vz0 = sh[5][cz][cy][cx];

  float fxc = 0.0f, fyc = 0.0f, fzc = 0.0f;
  float fxd = 0.0f, fyd = 0.0f, fzd = 0.0f;

  for (int i = 0; i < 5; ++i) {
    const int oi = 2 - i; // offset along axis2 (x, contiguous)
    for (int j = 0; j < 5; ++j) {
      const int oj = 2 - j; // axis1
      #pragma unroll
      for (int k = 0; k < 5; ++k) {
        const int ok = 2 - k; // axis0
        const int nz = cz + ok, ny = cy + oj, nx = cx + oi;
        const float dx = x0 - sh[0][nz][ny][nx];
        const float dy = y0 - sh[1][nz][ny][nx];
        const float dz = z0 - sh[2][nz][ny][nx];
        const float dist = sqrtf(dx * dx + dy * dy + dz * dz);
        const bool  ov   = dist < two_d;
        const float safe = fmaxf(EPS_C, dist);
        const float rs   = 1.0f / safe;
        const float coef = KN_C * (dist - two_d) * rs;
        fxc += ov ? coef * dx : 0.0f;
        fyc += ov ? coef * dy : 0.0f;
        fzc += ov ? coef * dz : 0.0f;
        const float dvx = vx0 - sh[3][nz][ny][nx];
        const float dvy = vy0 - sh[4][nz][ny][nx];
        const float dvz = vz0 - sh[5][nz][ny][nx];
        const float dvn = (dvx * dx + dvy * dy + dvz * dz) * rs;
        const float t   = eta * dvn * rs;
        fxd += ov ? t * dx : 0.0f;
        fyd += ov ? t * dy : 0.0f;
        fzd += ov ? t * dz : 0.0f;
      }
    }
  }

  // ---- friction: overwritten each shift in reference; only the final shift
  // (i=j=k=4 -> halo offsets -2,-2,-2) survives, using fully-accumulated f*.
  float frx, fry, frz;
  {
    const int nz = cz - 2, ny = cy - 2, nx = cx - 2;
    const float dx = x0 - sh[0][nz][ny][nx];
    const float dy = y0 - sh[1][nz][ny][nx];
    const float dz = z0 - sh[2][nz][ny][nx];
    const float dist = sqrtf(dx * dx + dy * dy + dz * dz);
    const bool  ov   = dist < two_d;
    const float dvx = vx0 - sh[3][nz][ny][nx];
    const float dvy = vy0 - sh[4][nz][ny][nx];
    const float dvz = vz0 - sh[5][nz][ny][nx];
    frx = -(ov ? fabsf(fabsf(MU_C * fyc) + fabsf(MU_C * fzc) - MU_C * fxd)
                   * dvx / fmaxf(EPS_C, fabsf(dvx)) : 0.0f);
    fry = -(ov ? fabsf(fabsf(MU_C * fxc) + fabsf(MU_C * fzc) - MU_C * fyd)
                   * dvy / fmaxf(EPS_C, fabsf(dvy)) : 0.0f);
    // NOTE: reference uses diffvy in the z-friction numerator — replicated.
    frz = -(ov ? fabsf(fabsf(MU_C * fxc) + fabsf(MU_C * fyc) - MU_C * fzd)
                   * dvy / fmaxf(EPS_C, fabsf(dvz)) : 0.0f);
  }

  const size_t NT  = (size_t)NN * NN * NN;
  const size_t idx = ((size_t)(bz + tz) * NN + (size_t)(by + ty)) * NN + (size_t)(bx + tx);
  out[0 * NT + idx] = fxc;
  out[1 * NT + idx] = fyc;
  out[2 * NT + idx] = fzc;
  out[3 * NT + idx] = fxd;
  out[4 * NT + idx] = fyd;
  out[5 * NT + idx] = fzd;
  out[6 * NT + idx] = frx;
  out[7 * NT + idx] = fry;
  out[8 * NT + idx] = frz;
}

extern "C" void kernel_launch(void* const* d_in, const int* in_sizes, int n_in,
                              void* d_out, int out_size, void* d_ws, size_t ws_size,
                              hipStream_t stream) {
  (void)in_sizes; (void)n_in; (void)out_size; (void)d_ws; (void)ws_size;
  const float* xg  = (const float*)d_in[0];
  const float* yg  = (const float*)d_in[1];
  const float* zg  = (const float*)d_in[2];
  const float* vxg = (const float*)d_in[3];
  const float* vyg = (const float*)d_in[4];
  const float* vzg = (const float*)d_in[5];
  const float* dp  = (const float*)d_in[6];
  float* out = (float*)d_out;

  // ETA computed exactly like the Python reference, in double precision.
  const double alpha = -log(0.7) / M_PI;
  const double gamma = alpha / sqrt(alpha * alpha + 1.0);
  const float  eta   = (float)(2.0 * gamma * sqrt(500000.0 * 1.0));

  dim3 block(TDX, TDY, TDZ);                    // 256 threads = 8 wave32
  dim3 grid(NN / TDX, NN / TDY, NN / TDZ);      // 12 x 48 x 48
  dem_forces_kernel<<<grid, block, 0, stream>>>(xg, yg, zg, vxg, vyg, vzg, dp, out, eta);
}